// SemanticIDQuantizer_71107478553160
// MI455X (gfx1250) — compile-verified
//
#include <hip/hip_runtime.h>
#include <hip/hip_bf16.h>

typedef __attribute__((ext_vector_type(2))) float v2f;
typedef __attribute__((ext_vector_type(8))) float v8f;

constexpr int B_ROWS = 8192;
constexpr int DDIM   = 512;
constexpr int KCB    = 2048;
constexpr int LLEV   = 4;
constexpr float LN_EPS = 1e-5f;

static __device__ __forceinline__ v8f wmma4(v2f a, v2f b, v8f c) {
  // D = A(16x4,f32) * B(4x16,f32) + C(16x16,f32)
  return __builtin_amdgcn_wmma_f32_16x16x4_f32(
      false, a, false, b, (short)0, c, false, false);
}

// ---------------------------------------------------------------------------
// Kernel 1: per-codebook-row squared norms ||cb_k||^2 for all L*K rows.
// One wave per row.
// ---------------------------------------------------------------------------
__global__ __launch_bounds__(256) void cbsq_kernel(
    const float* __restrict__ cbs, float* __restrict__ cbsq) {
  const int wave = threadIdx.x >> 5;
  const int lane = threadIdx.x & 31;
  const int row  = blockIdx.x * 8 + wave;            // 0 .. L*K-1
  const float* r = cbs + (size_t)row * DDIM;
  float s = 0.f;
  for (int j = lane; j < DDIM; j += 32) { float v = r[j]; s += v * v; }
  #pragma unroll
  for (int off = 16; off; off >>= 1) s += __shfl_xor(s, off, 32);
  if (lane == 0) cbsq[row] = s;
}

// ---------------------------------------------------------------------------
// Kernel 2: x = ReLU(LN_affine(F @ W^T + b)).  WMMA f32 16x16x4.
// Grid: B/16 workgroups of 256 threads (8 waves). Wave w owns 64 output cols.
// Full 16x512 tile staged in LDS for the LayerNorm.
// ---------------------------------------------------------------------------
__global__ __launch_bounds__(256) void proj_ln_relu_kernel(
    const float* __restrict__ F, const float* __restrict__ W,
    const float* __restrict__ bias, const float* __restrict__ g,
    const float* __restrict__ bln, float* __restrict__ X) {
  __shared__ float sy[16 * DDIM];
  __shared__ float smu[16];
  __shared__ float srs[16];

  const int tid  = threadIdx.x;
  const int wave = tid >> 5;
  const int lane = tid & 31;
  const int m0   = blockIdx.x * 16;
  const int half = lane >> 4;        // 0: K={0,1}, 1: K={2,3}
  const int l16  = lane & 15;
  const int n_base = wave * 64;

  const float* Arow = F + (size_t)(m0 + l16) * DDIM + half * 2;
  const float* Wrow = W + (size_t)(n_base + l16) * DDIM + half * 2;

  v8f z = {0.f,0.f,0.f,0.f,0.f,0.f,0.f,0.f};
  v8f acc0 = z, acc1 = z, acc2 = z, acc3 = z;

  #pragma unroll 4
  for (int kk = 0; kk < DDIM; kk += 4) {
    v2f a  = *(const v2f*)(Arow + kk);
    v2f b0 = *(const v2f*)(Wrow + kk);
    v2f b1 = *(const v2f*)(Wrow + 16 * DDIM + kk);
    v2f b2 = *(const v2f*)(Wrow + 32 * DDIM + kk);
    v2f b3 = *(const v2f*)(Wrow + 48 * DDIM + kk);
    acc0 = wmma4(a, b0, acc0);
    acc1 = wmma4(a, b1, acc1);
    acc2 = wmma4(a, b2, acc2);
    acc3 = wmma4(a, b3, acc3);
  }

  // C layout: VGPR r -> row r (lanes 0-15) / row r+8 (lanes 16-31), col = lane%16
  const int colb = n_base + l16;
  #pragma unroll
  for (int r = 0; r < 8; ++r) {
    const int row = r + half * 8;
    sy[row * DDIM + colb +  0] = acc0[r] + bias[colb +  0];
    sy[row * DDIM + colb + 16] = acc1[r] + bias[colb + 16];
    sy[row * DDIM + colb + 32] = acc2[r] + bias[colb + 32];
    sy[row * DDIM + colb + 48] = acc3[r] + bias[colb + 48];
  }
  __syncthreads();

  // LayerNorm stats: wave w handles rows 2w, 2w+1
  for (int rr = wave * 2; rr < wave * 2 + 2; ++rr) {
    float s = 0.f;
    for (int j = lane; j < DDIM; j += 32) s += sy[rr * DDIM + j];
    #pragma unroll
    for (int off = 16; off; off >>= 1) s += __shfl_xor(s, off, 32);
    const float mu = s * (1.0f / DDIM);
    float s2 = 0.f;
    for (int j = lane; j < DDIM; j += 32) {
      float d = sy[rr * DDIM + j] - mu; s2 += d * d;
    }
    #pragma unroll
    for (int off = 16; off; off >>= 1) s2 += __shfl_xor(s2, off, 32);
    if (lane == 0) { smu[rr] = mu; srs[rr] = rsqrtf(s2 * (1.0f / DDIM) + LN_EPS); }
  }
  __syncthreads();

  for (int idx = tid; idx < 16 * DDIM; idx += 256) {
    const int r = idx >> 9;
    const int c = idx & (DDIM - 1);
    float v = (sy[idx] - smu[r]) * srs[r];
    v = v * g[c] + bln[c];
    X[(size_t)(m0 + r) * DDIM + c] = fmaxf(v, 0.f);
  }
}

// ---------------------------------------------------------------------------
// Kernel 3 (per level): logits[b, lev, k] = -(||rs||^2 + ||cb_k||^2 - 2 rs.cb)/temp
// Grid: B/32 workgroups of 512 threads (16 waves). Each workgroup owns a
// 32-row x 2048-col tile; wave w owns 128 cols x both 16-row tiles, so every
// codebook (B) fragment load feeds two WMMAs and cb L2 traffic is halved
// versus a 16-row tile.
// ---------------------------------------------------------------------------
__global__ __launch_bounds__(512) void dist_logits_kernel(
    const float* __restrict__ residual, const float* __restrict__ cb,
    const float* __restrict__ cbsq, const float* __restrict__ scales,
    const float* __restrict__ temp_p, float* __restrict__ out, int lev) {
  __shared__ float rsq[32];

  const int tid  = threadIdx.x;
  const int wave = tid >> 5;       // 0..15
  const int lane = tid & 31;
  const int m0   = blockIdx.x * 32;
  const int half = lane >> 4;
  const int l16  = lane & 15;
  const float scale = scales[lev];
  const float invt  = 1.0f / fmaxf(temp_p[0], 0.01f);

  // ||rs||^2 per row: wave w reduces rows 2w and 2w+1
  for (int rr = wave * 2; rr < wave * 2 + 2; ++rr) {
    const float* rp = residual + (size_t)(m0 + rr) * DDIM;
    float s = 0.f;
    for (int j = lane; j < DDIM; j += 32) { float v = rp[j]; s += v * v; }
    #pragma unroll
    for (int off = 16; off; off >>= 1) s += __shfl_xor(s, off, 32);
    if (lane == 0) rsq[rr] = s * scale * scale;
  }
  __syncthreads();

  const int n_base = wave * 128;
  v8f acc0[8];   // rows m0 .. m0+15
  v8f acc1[8];   // rows m0+16 .. m0+31
  #pragma unroll
  for (int t = 0; t < 8; ++t) {
    acc0[t] = (v8f){0.f,0.f,0.f,0.f,0.f,0.f,0.f,0.f};
    acc1[t] = (v8f){0.f,0.f,0.f,0.f,0.f,0.f,0.f,0.f};
  }

  const float* A0row = residual + (size_t)(m0 + l16) * DDIM + half * 2;
  const float* A1row = A0row + (size_t)16 * DDIM;
  const float* Brow  = cb + (size_t)(n_base + l16) * DDIM + half * 2;

  for (int kk = 0; kk < DDIM; kk += 4) {
    v2f a0 = *(const v2f*)(A0row + kk);
    v2f a1 = *(const v2f*)(A1row + kk);
    a0.x *= scale; a0.y *= scale;        // rs = residual * scale
    a1.x *= scale; a1.y *= scale;
    #pragma unroll
    for (int t = 0; t < 8; ++t) {
      v2f b = *(const v2f*)(Brow + (size_t)t * 16 * DDIM + kk);
      acc0[t] = wmma4(a0, b, acc0[t]);
      acc1[t] = wmma4(a1, b, acc1[t]);
    }
  }

  #pragma unroll
  for (int t = 0; t < 8; ++t) {
    const int col = n_base + t * 16 + l16;
    const float cq = cbsq[col];
    #pragma unroll
    for (int r = 0; r < 8; ++r) {
      const int row0 = r + half * 8;
      const int row1 = row0 + 16;
      const float lg0 = (2.0f * acc0[t][r] - rsq[row0] - cq) * invt;
      const float lg1 = (2.0f * acc1[t][r] - rsq[row1] - cq) * invt;
      out[((size_t)(m0 + row0) * LLEV + lev) * KCB + col] = lg0;
      out[((size_t)(m0 + row1) * LLEV + lev) * KCB + col] = lg1;
    }
  }
}

// ---------------------------------------------------------------------------
// Kernel 4 (per level): argmax of logits per row (== argmin dist, first-index
// tie-break), gather codebook row, straight-through update of residual/qsum.
// One wave per row.
// ---------------------------------------------------------------------------
__global__ __launch_bounds__(256) void argmin_update_kernel(
    const float* __restrict__ logits, const float* __restrict__ cb,
    const float* __restrict__ scales, float* __restrict__ residual,
    float* __restrict__ qsum, int lev) {
  const int wave = threadIdx.x >> 5;
  const int lane = threadIdx.x & 31;
  const int row  = blockIdx.x * 8 + wave;
  const float scale = scales[lev];

  const float* lrow = logits + ((size_t)row * LLEV + lev) * KCB;
  float bestv = -3.402823466e38f;
  int   bestk = 0;
  for (int k = lane; k < KCB; k += 32) {
    float v = lrow[k];
    if (v > bestv) { bestv = v; bestk = k; }   // within lane k is increasing
  }
  #pragma unroll
  for (int off = 16; off; off >>= 1) {
    float ov = __shfl_xor(bestv, off, 32);
    int   ok = __shfl_xor(bestk, off, 32);
    if (ov > bestv || (ov == bestv && ok < bestk)) { bestv = ov; bestk = ok; }
  }

  const float* q = cb + (size_t)bestk * DDIM;
  float* rrow = residual + (size_t)row * DDIM;
  float* srow = qsum + (size_t)row * DDIM;
  for (int d = lane; d < DDIM; d += 32) {
    const float r0 = rrow[d];
    const float rs = r0 * scale;
    const float qe = q[d];
    const float quant = qe + (rs - qe);        // straight-through forward value
    srow[d] = (lev == 0) ? quant : (srow[d] + quant);
    rrow[d] = r0 - quant / scale;
  }
}

// ---------------------------------------------------------------------------
// Kernel 5: quantized_sum = LayerNorm(qsum) (no affine). One wave per row.
// ---------------------------------------------------------------------------
__global__ __launch_bounds__(256) void final_ln_kernel(
    const float* __restrict__ qsum, float* __restrict__ outq) {
  const int wave = threadIdx.x >> 5;
  const int lane = threadIdx.x & 31;
  const int row  = blockIdx.x * 8 + wave;
  const float* r = qsum + (size_t)row * DDIM;
  float s = 0.f;
  for (int j = lane; j < DDIM; j += 32) s += r[j];
  #pragma unroll
  for (int off = 16; off; off >>= 1) s += __shfl_xor(s, off, 32);
  const float mu = s * (1.0f / DDIM);
  float s2 = 0.f;
  for (int j = lane; j < DDIM; j += 32) { float d = r[j] - mu; s2 += d * d; }
  #pragma unroll
  for (int off = 16; off; off >>= 1) s2 += __shfl_xor(s2, off, 32);
  const float rsg = rsqrtf(s2 * (1.0f / DDIM) + LN_EPS);
  float* o = outq + (size_t)row * DDIM;
  for (int j = lane; j < DDIM; j += 32) o[j] = (r[j] - mu) * rsg;
}

// ---------------------------------------------------------------------------
extern "C" void kernel_launch(void* const* d_in, const int* in_sizes, int n_in,
                              void* d_out, int out_size, void* d_ws, size_t ws_size,
                              hipStream_t stream) {
  const float* F      = (const float*)d_in[0];   // [B, D]
  const float* W      = (const float*)d_in[1];   // [D, D]
  const float* bias   = (const float*)d_in[2];   // [D]
  const float* g      = (const float*)d_in[3];   // [D]
  const float* bln    = (const float*)d_in[4];   // [D]
  const float* cbs    = (const float*)d_in[5];   // [L, K, D]
  const float* scales = (const float*)d_in[6];   // [L]
  const float* temp   = (const float*)d_in[7];   // [1]
  float* out = (float*)d_out;                    // [B,L,K] logits ++ [B,D] qnorm

  float* residual = (float*)d_ws;                         // B*D floats
  float* qsum     = residual + (size_t)B_ROWS * DDIM;     // B*D floats
  float* cbsq     = qsum     + (size_t)B_ROWS * DDIM;     // L*K floats

  cbsq_kernel<<<(LLEV * KCB) / 8, 256, 0, stream>>>(cbs, cbsq);
  proj_ln_relu_kernel<<<B_ROWS / 16, 256, 0, stream>>>(F, W, bias, g, bln, residual);

  for (int l = 0; l < LLEV; ++l) {
    const float* cb_l = cbs + (size_t)l * KCB * DDIM;
    dist_logits_kernel<<<B_ROWS / 32, 512, 0, stream>>>(
        residual, cb_l, cbsq + (size_t)l * KCB, scales, temp, out, l);
    argmin_update_kernel<<<B_ROWS / 8, 256, 0, stream>>>(
        out, cb_l, scales, residual, qsum, l);
  }

  final_ln_kernel<<<B_ROWS / 8, 256, 0, stream>>>(
      qsum, out + (size_t)B_ROWS * LLEV * KCB);
}